// GCN3Conv_57724360458562
// MI455X (gfx1250) — compile-verified
//
#include <hip/hip_runtime.h>
#include <hip/hip_bf16.h>

// ---------------------------------------------------------------------------
// GCN3Conv for MI455X (gfx1250).
//  - fp32 WMMA GEMMs (v_wmma_f32_16x16x4_f32), W staged in LDS via the
//    gfx1250 async global->LDS path (global_load_async_to_lds_b128,
//    ASYNCcnt-tracked), B-fragments served from LDS (ds_load).
//  - Edge aggregation: wave-uniform edge ids scalarized via readfirstlane,
//    coalesced 512B gathers + f32 L2 atomics (feature matrix is L2-resident:
//    50000*128*4 = 25.6MB << 192MB L2).
// ---------------------------------------------------------------------------

typedef __attribute__((ext_vector_type(2))) float v2f;
typedef __attribute__((ext_vector_type(8))) float v8f;
typedef int gcn_v4i __attribute__((vector_size(16)));   // <4 x i32>

#if defined(__gfx1250__) && __has_builtin(__builtin_amdgcn_global_load_async_to_lds_b128)
#define GCN_ASYNC_LDS 1
#else
#define GCN_ASYNC_LDS 0
#endif

__global__ void gcn_fill_kernel(float* __restrict__ p, float v, int n) {
  int i = blockIdx.x * blockDim.x + threadIdx.x;
  if (i < n) p[i] = v;
}

__global__ void gcn_deg_accum_kernel(const int* __restrict__ dst,
                                     float* __restrict__ deg, int E) {
  int i = blockIdx.x * blockDim.x + threadIdx.x;
  if (i < E) atomicAdd(&deg[dst[i]], 1.0f);
}

__global__ void gcn_dinv_kernel(float* __restrict__ deg, int n) {
  int i = blockIdx.x * blockDim.x + threadIdx.x;
  if (i < n) deg[i] = rsqrtf(fmaxf(deg[i], 1e-12f));
}

// Small dense GEMM (K tiny or N-cols not 16-aligned): one thread per output.
__global__ void gcn_dense_small_kernel(const float* __restrict__ X,
                                       const float* __restrict__ W,
                                       const float* __restrict__ bias,
                                       float* __restrict__ Y,
                                       int M, int K, int Nc) {
  int gid = blockIdx.x * blockDim.x + threadIdx.x;
  if (gid >= M * Nc) return;
  int m = gid / Nc, j = gid - m * Nc;
  float s = bias ? bias[j] : 0.0f;
  for (int k = 0; k < K; ++k)
    s = fmaf(X[(size_t)m * K + k], W[(size_t)k * Nc + j], s);
  Y[gid] = s;
}

// WMMA fp32 GEMM: Y[M,Nc] = X[M,K] @ W[K,Nc] (+bias, optional relu).
// Requires K % 4 == 0, Nc % (16*NT) == 0, K*Nc <= 16384.
// Block = 8 waves; W is staged into LDS once per block (async if available).
// Each wave computes a 16 x (16*NT) strip: one A fragment feeds NT WMMAs.
// V_WMMA_F32_16X16X4_F32 fragment layout (ISA 7.12.2):
//   A: lanes 0-15 rows M=0..15 @ K={k0,k0+1}; lanes 16-31 @ K={k0+2,k0+3}
//   B: mirrored per-column; C/D: vgpr i -> M=i (lanes 0-15) / M=8+i (16-31)
template <int NT>
__global__ void gcn_wmma_gemm_kernel(const float* __restrict__ X,
                                     const float* __restrict__ W,
                                     const float* __restrict__ bias,
                                     float* __restrict__ Y,
                                     int M, int K, int Nc, int relu) {
  __shared__ float wlds[16384];  // 64KB: full W tile (K x Nc), row-major
  const int tid = threadIdx.y * 32 + threadIdx.x;  // 0..255
  const int KNc = K * Nc;

  // ---- stage W into LDS (16B per thread per iter) ----
  for (int idx = tid * 4; idx < KNc; idx += 256 * 4) {
#if GCN_ASYNC_LDS
    __builtin_amdgcn_global_load_async_to_lds_b128(
        (__attribute__((address_space(1))) gcn_v4i*)(W + idx),
        (__attribute__((address_space(3))) gcn_v4i*)(&wlds[idx]),
        /*offset=*/0, /*cpol=*/0);
#else
    *(float4*)(&wlds[idx]) = *(const float4*)(W + idx);
#endif
  }
#if GCN_ASYNC_LDS
#if __has_builtin(__builtin_amdgcn_s_wait_asynccnt)
  __builtin_amdgcn_s_wait_asynccnt(0);
#else
  asm volatile("s_wait_asynccnt 0" ::: "memory");
#endif
#endif
  __syncthreads();

  const int ngrp = Nc / (16 * NT);            // column groups per row-tile
  const int ntm  = (M + 15) >> 4;             // row tiles
  const int u    = blockIdx.x * blockDim.y + threadIdx.y;
  if (u >= ntm * ngrp) return;
  const int tm = u / ngrp;
  const int tg = u - tm * ngrp;

  const int lane = threadIdx.x;               // 0..31 (wave32)
  const int r    = lane & 15;
  const int kh   = (lane >> 4) << 1;          // 0 or 2
  const int row  = tm * 16 + r;
  const int rowc = row < M ? row : M - 1;     // clamp loads (store is guarded)
  const float* __restrict__ xrow = X + (size_t)rowc * K;
  const int colbase = tg * (16 * NT) + r;

  v8f acc[NT] = {};
  for (int k0 = 0; k0 < K; k0 += 4) {
    v2f a = *(const v2f*)(xrow + k0 + kh);    // K even -> 8B aligned
#pragma unroll
    for (int t = 0; t < NT; ++t) {
      const int col = colbase + t * 16;
      v2f b;
      b.x = wlds[(k0 + kh + 0) * Nc + col];   // ds_load, conflict-light
      b.y = wlds[(k0 + kh + 1) * Nc + col];
      acc[t] = __builtin_amdgcn_wmma_f32_16x16x4_f32(
          /*neg_a=*/false, a, /*neg_b=*/false, b,
          /*c_mod=*/(short)0, acc[t], /*reuse_a=*/false, /*reuse_b=*/false);
    }
  }

  const int mbase = tm * 16 + ((lane >> 4) ? 8 : 0);
#pragma unroll
  for (int t = 0; t < NT; ++t) {
    const int col = colbase + t * 16;
    const float bv = bias ? bias[col] : 0.0f;
#pragma unroll
    for (int i = 0; i < 8; ++i) {
      int mrow = mbase + i;
      if (mrow < M) {
        float v = acc[t][i] + bv;
        if (relu) v = fmaxf(v, 0.0f);
        Y[(size_t)mrow * Nc + col] = v;
      }
    }
  }
}

// y[n][f] = bias[f] + dinv[n]^2 * h[n][f]   (self-loop term folded in)
__global__ void gcn_init_y_kernel(const float* __restrict__ h,
                                  const float* __restrict__ dinv,
                                  const float* __restrict__ bias,
                                  float* __restrict__ y, int N, int H) {
  int gid = blockIdx.x * blockDim.x + threadIdx.x;
  if (gid >= N * H) return;
  int n = gid / H, f = gid - n * H;
  float di = dinv[n];
  y[gid] = bias[f] + di * di * h[gid];
}

// Edge scatter: block(32,8) -> one wave per edge; edge id is wave-uniform and
// forced scalar via readfirstlane so src/dst/dinv become scalar loads.
// Each lane handles 4 consecutive floats (float4 gather + 4 f32 atomics).
__global__ void gcn_edge_scatter_kernel(const float* __restrict__ h,
                                        const int* __restrict__ src,
                                        const int* __restrict__ dst,
                                        const float* __restrict__ dinv,
                                        float* __restrict__ y, int E, int H) {
  int e = blockIdx.x * blockDim.y + threadIdx.y;   // uniform within wave
  if (e >= E) return;
  e = __builtin_amdgcn_readfirstlane(e);
  const int s = src[e];
  const int d = dst[e];
  const float w = dinv[s] * dinv[d];
  const int f = threadIdx.x << 2;                  // lane*4; H mult of 128
  const float4 v = *(const float4*)(h + (size_t)s * H + f);
  float* yp = y + (size_t)d * H + f;
  atomicAdd(yp + 0, v.x * w);
  atomicAdd(yp + 1, v.y * w);
  atomicAdd(yp + 2, v.z * w);
  atomicAdd(yp + 3, v.w * w);
}

__global__ void gcn_relu_kernel(float* __restrict__ y, int n) {
  int i = blockIdx.x * blockDim.x + threadIdx.x;
  if (i < n) y[i] = fmaxf(y[i], 0.0f);
}

__global__ void gcn_pool_acc_kernel(const float* __restrict__ x,
                                    const int* __restrict__ batch,
                                    float* __restrict__ gsum,
                                    float* __restrict__ gcnt, int N, int H) {
  int gid = blockIdx.x * blockDim.x + threadIdx.x;
  if (gid >= N * H) return;
  int n = gid / H, f = gid - n * H;
  int b = batch[n];
  atomicAdd(&gsum[(size_t)b * H + f], x[gid]);
  if (f == 0) atomicAdd(&gcnt[b], 1.0f);
}

__global__ void gcn_mean_kernel(float* __restrict__ gsum,
                                const float* __restrict__ gcnt, int G, int H) {
  int gid = blockIdx.x * blockDim.x + threadIdx.x;
  if (gid >= G * H) return;
  int g = gid / H;
  gsum[gid] = gsum[gid] / fmaxf(gcnt[g], 1.0f);
}

static inline unsigned cdiv(long long a, long long b) {
  return (unsigned)((a + b - 1) / b);
}

extern "C" void kernel_launch(void* const* d_in, const int* in_sizes, int n_in,
                              void* d_out, int out_size, void* d_ws, size_t ws_size,
                              hipStream_t stream) {
  // Inputs (setup_inputs order)
  const float* pos  = (const float*)d_in[0];
  const int*   eidx = (const int*)d_in[1];     // [2,E] flat: src then dst
  const int*   batch= (const int*)d_in[2];
  const float* W1   = (const float*)d_in[3];
  const float* b1   = (const float*)d_in[4];
  const float* W2   = (const float*)d_in[5];
  const float* b2   = (const float*)d_in[6];
  const float* W3   = (const float*)d_in[7];
  const float* b3   = (const float*)d_in[8];
  const float* Wl1  = (const float*)d_in[9];
  const float* bl1  = (const float*)d_in[10];
  const float* Wl2  = (const float*)d_in[11];
  const float* bl2  = (const float*)d_in[12];

  const int D  = 3;
  const int N  = in_sizes[0] / D;
  const int E  = in_sizes[1] / 2;
  const int H  = in_sizes[4];            // 128
  const int Hh = in_sizes[10];           // 64
  const int C  = in_sizes[12];           // 10
  const int G  = out_size / C;           // 64

  const int* src = eidx;
  const int* dst = eidx + E;

  // Workspace carve-out (256B aligned slices)
  size_t off = 0;
  auto take = [&](size_t elems) -> float* {
    float* p = (float*)((char*)d_ws + off);
    off += ((elems * sizeof(float) + 255) / 256) * 256;
    return p;
  };
  float* dinv = take((size_t)N);            // degrees -> dinv (in place)
  float* bufH = take((size_t)N * H);        // GEMM output
  float* bufY = take((size_t)N * H);        // aggregated (layers 1,3)
  float* bufX = take((size_t)N * H);        // aggregated (layer 2)
  float* gsum = take((size_t)G * H);
  float* gcnt = take((size_t)G);
  float* h1   = take((size_t)G * Hh);
  (void)ws_size; (void)n_in;

  const int TB = 256;
  dim3 wblk(32, 8);                         // 8 waves / block

  // --- degrees (self-loop => init 1) and dinv = rsqrt(deg) ---
  gcn_fill_kernel<<<cdiv(N, TB), TB, 0, stream>>>(dinv, 1.0f, N);
  gcn_deg_accum_kernel<<<cdiv(E, TB), TB, 0, stream>>>(dst, dinv, E);
  gcn_dinv_kernel<<<cdiv(N, TB), TB, 0, stream>>>(dinv, N);

  auto gemm = [&](const float* X, const float* W, const float* bias,
                  float* Y, int M, int K, int Nc, int relu) {
    // NT=4: each wave computes a 16x64 strip; Nc must be a multiple of 64.
    int units = (((M + 15) / 16)) * (Nc / 64);
    gcn_wmma_gemm_kernel<4><<<cdiv(units, 8), wblk, 0, stream>>>(
        X, W, bias, Y, M, K, Nc, relu);
  };
  auto aggregate = [&](const float* h, const float* bias, float* y) {
    gcn_init_y_kernel<<<cdiv((long long)N * H, TB), TB, 0, stream>>>(
        h, dinv, bias, y, N, H);
    gcn_edge_scatter_kernel<<<cdiv(E, 8), wblk, 0, stream>>>(
        h, src, dst, dinv, y, E, H);
    gcn_relu_kernel<<<cdiv((long long)N * H, TB), TB, 0, stream>>>(
        y, N * H);
  };

  // --- layer 1: h = pos @ W1 (K=3, scalar path), aggregate -> bufY ---
  gcn_dense_small_kernel<<<cdiv((long long)N * H, TB), TB, 0, stream>>>(
      pos, W1, nullptr, bufH, N, D, H);
  aggregate(bufH, b1, bufY);

  // --- layer 2: WMMA GEMM, aggregate -> bufX ---
  gemm(bufY, W2, nullptr, bufH, N, H, H, 0);
  aggregate(bufH, b2, bufX);

  // --- layer 3: WMMA GEMM, aggregate -> bufY ---
  gemm(bufX, W3, nullptr, bufH, N, H, H, 0);
  aggregate(bufH, b3, bufY);

  // --- global mean pool ---
  gcn_fill_kernel<<<cdiv((long long)G * H, TB), TB, 0, stream>>>(gsum, 0.0f, G * H);
  gcn_fill_kernel<<<1, TB, 0, stream>>>(gcnt, 0.0f, G);
  gcn_pool_acc_kernel<<<cdiv((long long)N * H, TB), TB, 0, stream>>>(
      bufY, batch, gsum, gcnt, N, H);
  gcn_mean_kernel<<<cdiv((long long)G * H, TB), TB, 0, stream>>>(gsum, gcnt, G, H);

  // --- MLP head: h1 = relu(g @ Wl1 + bl1)  [64x128x64 -> WMMA] ---
  gemm(gsum, Wl1, bl1, h1, G, H, Hh, 1);
  // --- out = h1 @ Wl2 + bl2  [64x64x10 -> scalar path] ---
  gcn_dense_small_kernel<<<cdiv((long long)G * C, TB), TB, 0, stream>>>(
      h1, Wl2, bl2, (float*)d_out, G, Hh, C);
}